// Model_74268574483080
// MI455X (gfx1250) — compile-verified
//
#include <hip/hip_runtime.h>
#include <math.h>

typedef __attribute__((ext_vector_type(2))) float v2f;
typedef __attribute__((ext_vector_type(8))) float v8f;

#define N_TRAJ 6
#define TL 80
#define D_OUT 966            // N_TRAJ*2*TL + N_TRAJ
#define B_ROWS 4096
#define K_DIM 2048

#define BM 128
#define BN 64
#define KT 16
#define LDA 20               // padded LDS row stride (floats); multiple of 4 -> 16B alignment
#define LDB 20

__global__ __launch_bounds__(256)
void gemm_wmma_f32(const float* __restrict__ X, const float* __restrict__ W,
                   float* __restrict__ out)
{
  __shared__ float Als[2][BM * LDA];   // A tile, row-major [128][16] (+pad), double-buffered
  __shared__ float Bls[2][BN * LDB];   // B tile, TRANSPOSED [64][16] (+pad), double-buffered

  const int m0   = blockIdx.y * BM;
  const int n0   = blockIdx.x * BN;
  const int tid  = threadIdx.x;
  const int wid  = tid >> 5;
  const int lane = tid & 31;
  const int lo   = lane & 15;       // N (or M) position within 16
  const int hi   = lane >> 4;       // K half-select for A/B fragments
  const int wm   = (wid >> 1) * 32; // wave M offset in block tile
  const int wn   = (wid &  1) * 32; // wave N offset in block tile

  v8f acc00 = {}, acc01 = {}, acc10 = {}, acc11 = {};

  // ---- A staging map: thread -> (row ar, row ar+64) x float4 at col ac
  const int ar = tid >> 2;           // 0..63
  const int ac = (tid & 3) * 4;      // 0,4,8,12
  // ---- B staging map: thread -> 1 column, 4 consecutive k-rows (coalesced loads)
  const int bn  = tid & 63;          // 0..63 column within tile
  const int bkg = (tid >> 6) * 4;    // 0,4,8,12 k-row group
  const int  gcol   = n0 + bn;
  const bool colOk  = gcol < D_OUT;
  const int  gcolC  = colOk ? gcol : (D_OUT - 1);  // clamp (branch-free loads)
  const float bmask = colOk ? 1.0f : 0.0f;

  float4 aA, aB;
  float  b0, b1, b2, b3;

  auto global_stage = [&](int k0) {
    aA = *(const float4*)&X[(size_t)(m0 + ar)      * K_DIM + k0 + ac];
    aB = *(const float4*)&X[(size_t)(m0 + ar + 64) * K_DIM + k0 + ac];
    b0 = W[(size_t)(k0 + bkg + 0) * D_OUT + gcolC] * bmask;
    b1 = W[(size_t)(k0 + bkg + 1) * D_OUT + gcolC] * bmask;
    b2 = W[(size_t)(k0 + bkg + 2) * D_OUT + gcolC] * bmask;
    b3 = W[(size_t)(k0 + bkg + 3) * D_OUT + gcolC] * bmask;
  };

  auto lds_store = [&](int buf) {
    *(float4*)&Als[buf][(ar)      * LDA + ac] = aA;
    *(float4*)&Als[buf][(ar + 64) * LDA + ac] = aB;
    float4 bv; bv.x = b0; bv.y = b1; bv.z = b2; bv.w = b3;
    *(float4*)&Bls[buf][bn * LDB + bkg] = bv;   // transpose on store: [n][k]
  };

  auto compute = [&](int buf) {
#pragma unroll
    for (int k = 0; k < KT; k += 4) {
      // A 16x4 f32 fragment: lane lo = M, float2 = K {k+2hi, k+2hi+1}
      v2f a0  = *(const v2f*)&Als[buf][(wm +      lo) * LDA + k + 2 * hi];
      v2f a1  = *(const v2f*)&Als[buf][(wm + 16 + lo) * LDA + k + 2 * hi];
      // B 4x16 f32 fragment: lane lo = N, float2 = K {k+2hi, k+2hi+1}
      v2f bb0 = *(const v2f*)&Bls[buf][(wn +      lo) * LDB + k + 2 * hi];
      v2f bb1 = *(const v2f*)&Bls[buf][(wn + 16 + lo) * LDB + k + 2 * hi];
      acc00 = __builtin_amdgcn_wmma_f32_16x16x4_f32(false, a0, false, bb0,
                                                    (short)0, acc00, false, false);
      acc01 = __builtin_amdgcn_wmma_f32_16x16x4_f32(false, a0, false, bb1,
                                                    (short)0, acc01, false, false);
      acc10 = __builtin_amdgcn_wmma_f32_16x16x4_f32(false, a1, false, bb0,
                                                    (short)0, acc10, false, false);
      acc11 = __builtin_amdgcn_wmma_f32_16x16x4_f32(false, a1, false, bb1,
                                                    (short)0, acc11, false, false);
    }
  };

  // ---- software pipeline: double-buffered LDS, one barrier per K-chunk
  global_stage(0);
  lds_store(0);
  __syncthreads();
  int cur = 0;
  for (int k0 = KT; k0 < K_DIM; k0 += KT) {
    global_stage(k0);        // prefetch next tile (overlaps WMMA below)
    compute(cur);            // 16x v_wmma_f32_16x16x4_f32 from LDS buffer cur
    lds_store(cur ^ 1);      // waits on loads, fills other buffer
    __syncthreads();
    cur ^= 1;
  }
  compute(cur);

  // ---- scatter-store: cols [0,6) -> conf [4096][6]; cols [6,966) -> traj region
  float* conf    = out;                       // 4096*6 floats
  float* trajbuf = out + B_ROWS * N_TRAJ;     // 4096*960 floats (logits now, traj later)

  auto store_tile = [&](const v8f& a, int mi, int nj) {
    const int gn = n0 + wn + nj * 16 + lo;
    if (gn >= D_OUT) return;
    const int gmBase = m0 + wm + mi * 16 + hi * 8;   // C/D: lane-half = M halves
#pragma unroll
    for (int r = 0; r < 8; ++r) {
      const float v = a[r];
      const int gm = gmBase + r;
      if (gn < N_TRAJ)
        conf[(size_t)gm * N_TRAJ + gn] = v;
      else
        trajbuf[(size_t)gm * (N_TRAJ * 2 * TL) + (gn - N_TRAJ)] = v;
    }
  };
  store_tile(acc00, 0, 0);
  store_tile(acc01, 0, 1);
  store_tile(acc10, 1, 0);
  store_tile(acc11, 1, 1);
}

// One thread per (batch, trajectory): sequential 80-step bicycle model.
// Reads logits in-place from the traj region of d_out and overwrites with (px,py).
__global__ __launch_bounds__(256)
void integrate_kernel(float* __restrict__ out,
                      const float* __restrict__ length,
                      const float* __restrict__ velocity)
{
  const int idx = blockIdx.x * blockDim.x + threadIdx.x;   // b*N_TRAJ + traj
  if (idx >= B_ROWS * N_TRAJ) return;
  const int b = idx / N_TRAJ;

  const float MAX_STEER = 37.0f * 3.14159265358979323846f / 180.0f;
  const float MAX_ACC   = 2.001f;
  const float DT        = 0.1f;

  const float lr = length[b] * 0.24f;
  float v  = velocity[b];
  float h  = 0.0f, px = 0.0f, py = 0.0f;

  float* p = out + B_ROWS * N_TRAJ + (size_t)idx * (2 * TL);

  for (int t = 0; t < TL; ++t) {
    float steer = p[2 * t + 0];
    float acc   = p[2 * t + 1];
    steer = fminf(fmaxf(steer, -MAX_STEER), MAX_STEER);
    acc   = fminf(fmaxf(acc,   -MAX_ACC),   MAX_ACC);
    const float beta = atanf(0.4f * tanf(steer));
    v += acc * DT;
    h += (v / lr) * sinf(beta) * DT;
    const float hb = h + beta;
    px += v * cosf(hb) * DT;
    py += v * sinf(hb) * DT;
    p[2 * t + 0] = px;          // overwrite logits in place with trajectory
    p[2 * t + 1] = py;
  }
}

extern "C" void kernel_launch(void* const* d_in, const int* in_sizes, int n_in,
                              void* d_out, int out_size, void* d_ws, size_t ws_size,
                              hipStream_t stream) {
  const float* X        = (const float*)d_in[0];   // [4096, 2048]
  const float* W        = (const float*)d_in[1];   // [2048, 966]
  const float* length   = (const float*)d_in[2];   // [4096]
  const float* velocity = (const float*)d_in[3];   // [4096]
  float* out = (float*)d_out;                      // conf(24576) ++ traj(3932160)

  dim3 gblocks((D_OUT + BN - 1) / BN, B_ROWS / BM);   // (16, 32)
  gemm_wmma_f32<<<gblocks, 256, 0, stream>>>(X, W, out);

  const int total = B_ROWS * N_TRAJ;
  integrate_kernel<<<(total + 255) / 256, 256, 0, stream>>>(out, length, velocity);
}